// VoronoiHeatModel_63677185130780
// MI455X (gfx1250) — compile-verified
//
#include <hip/hip_runtime.h>

typedef __attribute__((ext_vector_type(2))) float v2f;
typedef __attribute__((ext_vector_type(8))) float v8f;

#define NSEEDS 16
#define NITERS 40

// ---------------------------------------------------------------- utilities
__global__ void k_zero(float* __restrict__ p, int n) {
  int i = blockIdx.x * blockDim.x + threadIdx.x;
  if (i < n) p[i] = 0.f;
}

// L_diag[v] = sum of val where row==col==v
__global__ void k_ldiag(const int* __restrict__ row, const int* __restrict__ col,
                        const float* __restrict__ val, float* __restrict__ Ldiag, int nnz) {
  int e = blockIdx.x * blockDim.x + threadIdx.x;
  if (e < nnz) {
    int r = row[e];
    if (r == col[e]) atomicAdd(&Ldiag[r], val[e]);
  }
}

// Minv[v] = 1 / max(M + t*Ldiag, 1e-12)
__global__ void k_precond(const float* __restrict__ M, const float* __restrict__ Ldiag,
                          const float* __restrict__ tptr, float* __restrict__ Minv, int nV) {
  int v = blockIdx.x * blockDim.x + threadIdx.x;
  if (v < nV) {
    float A = M[v] + tptr[0] * Ldiag[v];
    Minv[v] = 1.f / fmaxf(A, 1e-12f);
  }
}

// X=0, R=B, P=Minv*B, rz[s] += sum(R*Z)  (grid-stride keeps s = tid&15 fixed)
__global__ void k_cg_init(const float* __restrict__ Bm, const float* __restrict__ Minv,
                          float* __restrict__ X, float* __restrict__ R, float* __restrict__ P,
                          float* __restrict__ rz, int n16) {
  __shared__ float acc[NSEEDS];
  int t = threadIdx.x;
  if (t < NSEEDS) acc[t] = 0.f;
  __syncthreads();
  float local = 0.f;
  int stride = gridDim.x * blockDim.x;           // multiple of 16
  for (int idx = blockIdx.x * blockDim.x + t; idx < n16; idx += stride) {
    float r = Bm[idx];
    float z = Minv[idx >> 4] * r;
    X[idx] = 0.f; R[idx] = r; P[idx] = z;
    local += r * z;
  }
  atomicAdd(&acc[t & 15], local);
  __syncthreads();
  if (t < NSEEDS) atomicAdd(&rz[t], acc[t]);
}

// AP = M_diag * P
__global__ void k_ap_init(const float* __restrict__ Mdiag, const float* __restrict__ P,
                          float* __restrict__ AP, int n16) {
  int idx = blockIdx.x * blockDim.x + threadIdx.x;
  if (idx < n16) AP[idx] = Mdiag[idx >> 4] * P[idx];
}

// AP[row,:] += t*val * P[col,:]   (one thread per nnz, 16 seeds via float4 loads)
__global__ void k_spmv_scatter(const int* __restrict__ row, const int* __restrict__ col,
                               const float* __restrict__ val, const float* __restrict__ tptr,
                               const float* __restrict__ P, float* __restrict__ AP, int nnz) {
  int e = blockIdx.x * blockDim.x + threadIdx.x;
  if (e >= nnz) return;
  float w = tptr[0] * val[e];
  int r = row[e] << 4, c = col[e] << 4;
  const float4* pc = (const float4*)(P + c);
  float4 p0 = pc[0], p1 = pc[1], p2 = pc[2], p3 = pc[3];
  float* ap = AP + r;
  atomicAdd(ap + 0,  w * p0.x); atomicAdd(ap + 1,  w * p0.y);
  atomicAdd(ap + 2,  w * p0.z); atomicAdd(ap + 3,  w * p0.w);
  atomicAdd(ap + 4,  w * p1.x); atomicAdd(ap + 5,  w * p1.y);
  atomicAdd(ap + 6,  w * p1.z); atomicAdd(ap + 7,  w * p1.w);
  atomicAdd(ap + 8,  w * p2.x); atomicAdd(ap + 9,  w * p2.y);
  atomicAdd(ap + 10, w * p2.z); atomicAdd(ap + 11, w * p2.w);
  atomicAdd(ap + 12, w * p3.x); atomicAdd(ap + 13, w * p3.y);
  atomicAdd(ap + 14, w * p3.z); atomicAdd(ap + 15, w * p3.w);
}

// denom[s] += sum_v P*AP
__global__ void k_dot(const float* __restrict__ a, const float* __restrict__ b,
                      float* __restrict__ out16, int n16) {
  __shared__ float acc[NSEEDS];
  int t = threadIdx.x;
  if (t < NSEEDS) acc[t] = 0.f;
  __syncthreads();
  float local = 0.f;
  int stride = gridDim.x * blockDim.x;
  for (int idx = blockIdx.x * blockDim.x + t; idx < n16; idx += stride)
    local += a[idx] * b[idx];
  atomicAdd(&acc[t & 15], local);
  __syncthreads();
  if (t < NSEEDS) atomicAdd(&out16[t], acc[t]);
}

// X += alpha*P; R -= alpha*AP; rz_new[s] += R * (Minv*R)
__global__ void k_update_xr(float* __restrict__ X, float* __restrict__ R,
                            const float* __restrict__ P, const float* __restrict__ AP,
                            const float* __restrict__ Minv, const float* __restrict__ rzc,
                            const float* __restrict__ denom, float* __restrict__ rzn, int n16) {
  __shared__ float acc[NSEEDS];
  int t = threadIdx.x;
  if (t < NSEEDS) acc[t] = 0.f;
  __syncthreads();
  int s = t & 15;
  float alpha = rzc[s] / fmaxf(denom[s], 1e-30f);
  float local = 0.f;
  int stride = gridDim.x * blockDim.x;
  for (int idx = blockIdx.x * blockDim.x + t; idx < n16; idx += stride) {
    float x = X[idx] + alpha * P[idx];
    float r = R[idx] - alpha * AP[idx];
    X[idx] = x; R[idx] = r;
    local += r * (Minv[idx >> 4] * r);
  }
  atomicAdd(&acc[s], local);
  __syncthreads();
  if (t < NSEEDS) atomicAdd(&rzn[t], acc[t]);
}

// P = Minv*R + beta*P
__global__ void k_update_p(float* __restrict__ P, const float* __restrict__ R,
                           const float* __restrict__ Minv, const float* __restrict__ rzc,
                           const float* __restrict__ rzn, int n16) {
  int idx = blockIdx.x * blockDim.x + threadIdx.x;
  if (idx < n16) {
    int s = idx & 15;
    float beta = rzn[s] / fmaxf(rzc[s], 1e-30f);
    P[idx] = Minv[idx >> 4] * R[idx] + beta * P[idx];
  }
}

// U = X ; S = -log(max(nan_to_num(U), 1e-9))
__global__ void k_outputs(const float* __restrict__ X, float* __restrict__ Uo,
                          float* __restrict__ So, int n16) {
  int idx = blockIdx.x * blockDim.x + threadIdx.x;
  if (idx < n16) {
    float u = X[idx];
    Uo[idx] = u;
    unsigned ub  = __float_as_uint(u);
    unsigned mag = ub & 0x7fffffffu;
    float us = u;
    if (mag > 0x7f800000u)       us = 1e-9f;                       // NaN
    else if (mag == 0x7f800000u) us = (ub >> 31) ? 0.0f : 1.0f;    // -Inf / +Inf
    So[idx] = -__logf(fmaxf(us, 1e-9f));
  }
}

// One wave per face: grad(16x3) = A(16x4: uI uJ uK 0) x B(4x16: gI gJ gK 0 in cols 0..2)
// via V_WMMA_F32_16X16X4_F32, then normalize and write Xdir[f,16,3].
// ISA A layout 16x4 f32: lanes0-15 -> K0(v0),K1(v1); lanes16-31 -> K2(v0),K3(v1).
// B layout 4x16 f32 (row striped across lanes): lanes0-15 -> rows K0/K1; lanes16-31 -> K2/K3.
// D layout: VGPR v, lanes0-15 M=v, lanes16-31 M=v+8; N = lane%16.
__global__ void k_grad_wmma(const int* __restrict__ Fv, const float* __restrict__ X,
                            const float* __restrict__ gI, const float* __restrict__ gJ,
                            const float* __restrict__ gK, float* __restrict__ xdir, int nF) {
  int face = blockIdx.x * (blockDim.x >> 5) + (threadIdx.x >> 5); // wave-uniform
  if (face >= nF) return;
  int lane = threadIdx.x & 31;
  int half = lane >> 4;
  int m = lane & 15;                    // seed index (A rows) / dim column (B,D cols)

  int i0 = Fv[face * 3 + 0], i1 = Fv[face * 3 + 1], i2 = Fv[face * 3 + 2];
  // branch-free loads so EXEC stays all-1s into the WMMA
  float u0 = X[i0 * 16 + m], u1 = X[i1 * 16 + m], u2 = X[i2 * 16 + m];
  v2f a;
  a.x = half ? u2  : u0;
  a.y = half ? 0.f : u1;

  int nc = (m < 3) ? m : 0;
  float msk = (m < 3) ? 1.f : 0.f;
  float bI = gI[face * 3 + nc] * msk;
  float bJ = gJ[face * 3 + nc] * msk;
  float bK = gK[face * 3 + nc] * msk;
  v2f b;
  b.x = half ? bK  : bI;
  b.y = half ? 0.f : bJ;

  v8f c = {};
  c = __builtin_amdgcn_wmma_f32_16x16x4_f32(false, a, false, b, (short)0, c, false, false);

  #pragma unroll
  for (int v = 0; v < 8; ++v) {
    float g  = c[v];                            // grad[seed = v+8*half][dim = m]
    float g0 = __shfl(g, half * 16 + 0, 32);
    float g1 = __shfl(g, half * 16 + 1, 32);
    float g2 = __shfl(g, half * 16 + 2, 32);
    float nrm = fmaxf(__fsqrt_rn(g0 * g0 + g1 * g1 + g2 * g2), 1e-12f);
    if (m < 3) {
      int s = v + 8 * half;
      xdir[(size_t)face * 48 + s * 3 + m] = -g / nrm;
    }
  }
}

// ---------------------------------------------------------------- driver
extern "C" void kernel_launch(void* const* d_in, const int* in_sizes, int n_in,
                              void* d_out, int out_size, void* d_ws, size_t ws_size,
                              hipStream_t stream) {
  const int*   F    = (const int*)d_in[0];
  const int*   row  = (const int*)d_in[1];
  const int*   col  = (const int*)d_in[2];
  const float* val  = (const float*)d_in[3];
  const float* M    = (const float*)d_in[4];
  const float* gI   = (const float*)d_in[5];
  const float* gJ   = (const float*)d_in[6];
  const float* gK   = (const float*)d_in[7];
  const float* Bm   = (const float*)d_in[8];
  const float* tptr = (const float*)d_in[9];

  int nF  = in_sizes[0] / 3;
  int nnz = in_sizes[1];
  int nV  = in_sizes[4];
  int n16 = nV * NSEEDS;

  float* ws    = (float*)d_ws;
  float* X     = ws;
  float* R     = X + n16;
  float* P     = R + n16;
  float* AP    = P + n16;
  float* Minv  = AP + n16;
  float* Ldiag = Minv + nV;
  float* scal  = Ldiag + nV;       // [0:16]=denom, [16:32]=rzA, [32:48]=rzB
  float* denom = scal;
  float* rzbuf = scal + 16;

  const int T = 256;
  int gV   = (nV  + T - 1) / T;
  int gN16 = (n16 + T - 1) / T;
  int gE   = (nnz + T - 1) / T;
  const int RB = 256;              // reduction grid (stride multiple of 16)

  k_zero<<<gV, T, 0, stream>>>(Ldiag, nV);
  k_zero<<<1, 64, 0, stream>>>(scal, 48);
  k_ldiag<<<gE, T, 0, stream>>>(row, col, val, Ldiag, nnz);
  k_precond<<<gV, T, 0, stream>>>(M, Ldiag, tptr, Minv, nV);
  k_cg_init<<<RB, T, 0, stream>>>(Bm, Minv, X, R, P, rzbuf, n16);

  for (int it = 0; it < NITERS; ++it) {
    float* rzc = rzbuf + 16 * (it & 1);
    float* rzn = rzbuf + 16 * ((it + 1) & 1);
    k_zero<<<1, 32, 0, stream>>>(denom, 16);
    k_zero<<<1, 32, 0, stream>>>(rzn, 16);
    k_ap_init<<<gN16, T, 0, stream>>>(M, P, AP, n16);
    k_spmv_scatter<<<gE, T, 0, stream>>>(row, col, val, tptr, P, AP, nnz);
    k_dot<<<RB, T, 0, stream>>>(P, AP, denom, n16);
    k_update_xr<<<RB, T, 0, stream>>>(X, R, P, AP, Minv, rzc, denom, rzn, n16);
    k_update_p<<<gN16, T, 0, stream>>>(P, R, Minv, rzc, rzn, n16);
  }

  float* Uo = (float*)d_out;
  float* Xd = Uo + n16;
  float* So = Xd + (size_t)nF * 48;
  k_outputs<<<gN16, T, 0, stream>>>(X, Uo, So, n16);

  int gF = (nF + 7) / 8;           // 8 waves per 256-thread block, 1 face/wave
  k_grad_wmma<<<gF, 256, 0, stream>>>(F, X, gI, gJ, gK, Xd, nF);
}